// PoseLSTM_19207093747828
// MI455X (gfx1250) — compile-verified
//
#include <hip/hip_runtime.h>
#include <cmath>

// CDNA5 / gfx1250 fp32 WMMA fragments
typedef float v2f __attribute__((ext_vector_type(2)));
typedef float v8f __attribute__((ext_vector_type(8)));

#define T_DIM 1024
#define B_DIM 1024
#define L_DIM 10
#define H_DIM 10
#define G_DIM 40   // 4*H gates
#define K_DIM 20   // I + H (combined A = [x_t | h])
#define NT    3    // n-tiles of 16 (48 >= 40 gate columns, pad with zeros)
#define KS    5    // k-steps of 4 (covers K=20)

#define LOG2E_F  1.4426950408889634f

// Branch-free transcendentals on the HW v_exp_f32 / v_rcp_f32 units.
// Both saturate correctly for |x| large (exp2 -> inf, rcp(inf) -> 0).
__device__ __forceinline__ float sigmoid_fast(float x) {
    return __builtin_amdgcn_rcpf(1.0f + __builtin_amdgcn_exp2f(-x * LOG2E_F));
}
__device__ __forceinline__ float tanh_fast(float x) {
    const float e = __builtin_amdgcn_exp2f(x * (2.0f * LOG2E_F));
    return 1.0f - 2.0f * __builtin_amdgcn_rcpf(e + 1.0f);
}

// One wave (32 lanes) owns 16 batch rows and runs the entire L x T recursion.
// Weights live in VGPRs as WMMA B-fragments; h/c state lives in registers.
// Gate columns are permuted to n = 4*j + gate so i,f,g,o of hidden unit j
// are contiguous in the LDS gate buffer (one ds_load_b128 in the pointwise phase).
__global__ void __launch_bounds__(32)
lstm_stack_wmma_kernel(const float* __restrict__ x,
                       const float* __restrict__ hp,
                       const float* __restrict__ cp,
                       const float* __restrict__ W_ih,
                       const float* __restrict__ W_hh,
                       const float* __restrict__ b_ih,
                       const float* __restrict__ b_hh,
                       float* __restrict__ out)
{
    __shared__ __align__(16) float abuf[16 * K_DIM]; // [m][k]: cols 0..9 = x_t, 10..19 = h
    __shared__ __align__(16) float gbuf[16 * 48];    // [m][n]: gates, n = 4*j + gate

    const int lane = threadIdx.x;      // 0..31, wave32
    const int half = lane >> 4;        // WMMA half-wave
    const int lan  = lane & 15;
    const int b0   = blockIdx.x * 16;  // batch base for this wave

    float* ys  = out;                                        // [T][B][H], reused by every layer
    float* hns = out + (size_t)T_DIM * B_DIM * H_DIM;        // [L][B][H]
    float* cns = hns + (size_t)L_DIM * B_DIM * H_DIM;        // [L][B][H]

    // (m,j) pairs owned by this lane: p = lane + 32q, p = m*10 + j  (160 pairs)
    int pm[5], pj[5];
    #pragma unroll
    for (int q = 0; q < 5; ++q) {
        int p = lane + 32 * q;
        pm[q] = p / 10;
        pj[q] = p - 10 * pm[q];
    }

    for (int l = 0; l < L_DIM; ++l) {
        // ---- Build per-layer WMMA B-fragments (weights) and bias accumulator.
        // Combined B[k][n], k<10 -> W_ih, k>=10 -> W_hh; column n -> orig row g*10+j
        // (j = n>>2, g = n&3).  Columns n >= 40 are zero padding.
        // Assumed B layout (mirrors ISA A layout for 16x16x4 f32):
        //   lanes 0-15 : vgpr0 = B[k0+0][n], vgpr1 = B[k0+1][n]
        //   lanes 16-31: vgpr0 = B[k0+2][n], vgpr1 = B[k0+3][n]
        v2f bfrag[KS][NT];
        v8f cbias[NT];
        #pragma unroll
        for (int nt = 0; nt < NT; ++nt) {
            const int n = nt * 16 + lan;
            const bool valid = (n < G_DIM);
            int row = 0;
            float bv = 0.0f;
            if (valid) {
                const int j = n >> 2, g = n & 3;
                row = g * H_DIM + j;
                bv = b_ih[l * G_DIM + row] + b_hh[l * G_DIM + row];
            }
            #pragma unroll
            for (int s = 0; s < KS; ++s) {
                const int k0 = 4 * s + 2 * half;
                float w0 = 0.0f, w1 = 0.0f;
                if (valid) {
                    w0 = (k0 < H_DIM)
                         ? W_ih[(l * G_DIM + row) * H_DIM + k0]
                         : W_hh[(l * G_DIM + row) * H_DIM + (k0 - H_DIM)];
                    const int k1 = k0 + 1;
                    w1 = (k1 < H_DIM)
                         ? W_ih[(l * G_DIM + row) * H_DIM + k1]
                         : W_hh[(l * G_DIM + row) * H_DIM + (k1 - H_DIM)];
                }
                bfrag[s][nt][0] = w0;
                bfrag[s][nt][1] = w1;
            }
            v8f cb;
            #pragma unroll
            for (int r = 0; r < 8; ++r) cb[r] = bv; // bias independent of batch row
            cbias[nt] = cb;
        }

        // ---- Initial h/c state (coalesced: offset = l*B*H + b0*H + p)
        float hreg[5], creg[5];
        #pragma unroll
        for (int q = 0; q < 5; ++q) {
            const int off = l * B_DIM * H_DIM + b0 * H_DIM + lane + 32 * q;
            hreg[q] = hp[off];
            creg[q] = cp[off];
            abuf[pm[q] * K_DIM + H_DIM + pj[q]] = hreg[q];
        }

        const float* inp = (l == 0) ? x : ys;  // in-place ys: read own rows before write

        for (int t = 0; t < T_DIM; ++t) {
            const int tb = t * B_DIM * H_DIM + b0 * H_DIM;

            // Stage x_t rows into abuf cols [0,10): 5 coalesced dword loads/wave
            #pragma unroll
            for (int q = 0; q < 5; ++q) {
                const float v = inp[tb + lane + 32 * q];
                abuf[pm[q] * K_DIM + pj[q]] = v;
            }
            __syncthreads(); // single-wave WG: ordering fence only

            // A-fragments: A[m][k], m = lan for both halves, k0 = 4s + 2*half
            v2f afrag[KS];
            #pragma unroll
            for (int s = 0; s < KS; ++s) {
                const int k0 = 4 * s + 2 * half;
                afrag[s][0] = abuf[lan * K_DIM + k0];
                afrag[s][1] = abuf[lan * K_DIM + k0 + 1];
            }

            // G[16x40] = A[16x20] * B[20x40] + bias : 15x v_wmma_f32_16x16x4_f32
            #pragma unroll
            for (int nt = 0; nt < NT; ++nt) {
                v8f acc = cbias[nt];
                #pragma unroll
                for (int s = 0; s < KS; ++s) {
                    acc = __builtin_amdgcn_wmma_f32_16x16x4_f32(
                        false, afrag[s], false, bfrag[s][nt],
                        (short)0, acc, false, false);
                }
                #pragma unroll
                for (int r = 0; r < 8; ++r) {
                    const int mD = r + 8 * half;          // D: lanes 16-31 hold M=r+8
                    gbuf[mD * 48 + nt * 16 + lan] = acc[r];
                }
            }
            __syncthreads();

            // Pointwise LSTM update: 5 (m,j) pairs per lane, gates contiguous (b128).
            // Branch-free sigmoid/tanh keep the serial chain free of EXEC-mask
            // slow paths (pure v_exp_f32 / v_rcp_f32 / FMA sequences).
            #pragma unroll
            for (int q = 0; q < 5; ++q) {
                const float4 g4 = *(const float4*)&gbuf[pm[q] * 48 + 4 * pj[q]];
                const float ig = sigmoid_fast(g4.x);
                const float fg = sigmoid_fast(g4.y);
                const float gg = tanh_fast(g4.z);
                const float og = sigmoid_fast(g4.w);
                const float c  = fg * creg[q] + ig * gg;
                const float h  = og * tanh_fast(c);
                creg[q] = c;
                hreg[q] = h;
                abuf[pm[q] * K_DIM + H_DIM + pj[q]] = h;   // h for next timestep's A
                ys[tb + lane + 32 * q] = h;                // coalesced layer output
            }
            __syncthreads();
        }

        // ---- Final h/c for this layer (coalesced)
        #pragma unroll
        for (int q = 0; q < 5; ++q) {
            const int off = l * B_DIM * H_DIM + b0 * H_DIM + lane + 32 * q;
            hns[off] = hreg[q];
            cns[off] = creg[q];
        }
    }
}

extern "C" void kernel_launch(void* const* d_in, const int* in_sizes, int n_in,
                              void* d_out, int out_size, void* d_ws, size_t ws_size,
                              hipStream_t stream) {
    const float* x    = (const float*)d_in[0];
    const float* hp   = (const float*)d_in[1];
    const float* cp   = (const float*)d_in[2];
    const float* W_ih = (const float*)d_in[3];
    const float* W_hh = (const float*)d_in[4];
    const float* b_ih = (const float*)d_in[5];
    const float* b_hh = (const float*)d_in[6];
    float* out = (float*)d_out;
    (void)in_sizes; (void)n_in; (void)out_size; (void)d_ws; (void)ws_size;

    // 64 independent waves: one wave (32 threads) per 16 batch rows
    lstm_stack_wmma_kernel<<<B_DIM / 16, 32, 0, stream>>>(
        x, hp, cp, W_ih, W_hh, b_ih, b_hh, out);
}